// GCNLayer_80857054315142
// MI455X (gfx1250) — compile-verified
//
#include <hip/hip_runtime.h>

typedef __attribute__((ext_vector_type(2))) float v2f;
typedef __attribute__((ext_vector_type(8))) float v8f;

#define D 128

// ---------------- degree / normalization ----------------

__global__ __launch_bounds__(256) void gcn_deg_init(int* __restrict__ deg, int n) {
  int i = blockIdx.x * 256 + threadIdx.x;
  if (i < n) deg[i] = 1;  // self-loop contribution
}

__global__ __launch_bounds__(256) void gcn_deg_accum(const int* __restrict__ dst,
                                                     int* __restrict__ deg, int e) {
  int i = blockIdx.x * 256 + threadIdx.x;
  if (i < e) atomicAdd(&deg[dst[i]], 1);
}

__global__ __launch_bounds__(256) void gcn_dinv(const int* __restrict__ deg,
                                                float* __restrict__ dinv, int n) {
  int i = blockIdx.x * 256 + threadIdx.x;
  if (i < n) dinv[i] = rsqrtf((float)deg[i]);  // deg >= 1 always (self loops)
}

// ---------------- h = x @ W  (fp32 WMMA) ----------------
// Block = 256 threads = 8 wave32s. Block computes 16 rows x 128 cols of h.
// Wave w owns the 16x16 tile at colBase = 16*w. K=128 swept in steps of 4
// with V_WMMA_F32_16X16X4_F32. W (64KB) and the 16x128 x-tile live in LDS.
__global__ __launch_bounds__(256) void gcn_gemm(const float* __restrict__ x,
                                                const float* __restrict__ W,
                                                float* __restrict__ h, int n) {
  __shared__ float sA[16 * 132];  // x tile, padded stride 132 dwords (bank spread)
  __shared__ float sB[D * D];     // full W, rows contiguous

  const int tid = threadIdx.x;
  const int rowBase = blockIdx.x * 16;

  // cooperative load W: 4096 float4 chunks over 256 threads
  for (int j = tid; j < (D * D) / 4; j += 256) {
    *(float4*)&sB[j * 4] = *(const float4*)&W[j * 4];
  }
  // cooperative load x tile: 512 float4 chunks (32 chunks per 128-wide row)
  for (int j = tid; j < (16 * D) / 4; j += 256) {
    int r = j >> 5;
    int c = (j & 31) * 4;
    float4 v = make_float4(0.f, 0.f, 0.f, 0.f);
    if (rowBase + r < n) v = *(const float4*)&x[(size_t)(rowBase + r) * D + c];
    *(float4*)&sA[r * 132 + c] = v;
  }
  __syncthreads();

  const int wave = tid >> 5;
  const int lane = tid & 31;
  const int hh   = lane >> 4;   // lane half: K pair select
  const int lrow = lane & 15;
  const int colBase = wave * 16;

  v8f acc = {};
#pragma unroll
  for (int k0 = 0; k0 < D; k0 += 4) {
    // A 16x4 f32: v0 = K=0 (lanes 0-15) / K=2 (16-31); v1 = K=1 / K=3
    v2f a, b;
    a.x = sA[lrow * 132 + k0 + 2 * hh];
    a.y = sA[lrow * 132 + k0 + 2 * hh + 1];
    // B 4x16 f32: row (K) striped across vgprs, N across lanes
    b.x = sB[(k0 + 2 * hh) * D + colBase + lrow];
    b.y = sB[(k0 + 2 * hh + 1) * D + colBase + lrow];
    acc = __builtin_amdgcn_wmma_f32_16x16x4_f32(
        /*neg_a=*/false, a, /*neg_b=*/false, b,
        /*c_mod=*/(short)0, acc, /*reuse_a=*/false, /*reuse_b=*/false);
  }

  // C/D layout: vgpr i -> M = i (lanes 0-15) / i+8 (lanes 16-31), N = lrow
#pragma unroll
  for (int i = 0; i < 8; ++i) {
    int r = rowBase + i + 8 * hh;
    if (r < n) h[(size_t)r * D + colBase + lrow] = acc[i];
  }
}

// ---------------- out = b + dinv^2 * h  (self-loop term, plain stores) ----------------
__global__ __launch_bounds__(256) void gcn_out_init(const float* __restrict__ h,
                                                    const float* __restrict__ dinv,
                                                    const float* __restrict__ b,
                                                    float* __restrict__ out, int n) {
  int i = blockIdx.x * 256 + threadIdx.x;  // one lane per 4 channels
  int node = i >> 5;
  int lane = i & 31;
  if (node >= n) return;
  float di = dinv[node];
  float c = di * di;
  float4 hv = *(const float4*)&h[(size_t)node * D + lane * 4];
  float4 bv = *(const float4*)&b[lane * 4];
  float4 o = make_float4(bv.x + c * hv.x, bv.y + c * hv.y,
                         bv.z + c * hv.z, bv.w + c * hv.w);
  *(float4*)&out[(size_t)node * D + lane * 4] = o;
}

// ---------------- edge scatter: out[dst] += dinv[src]*dinv[dst] * h[src] ----------------
// One wave32 per edge, 4 channels per lane. h/out are L2-resident (51.2MB each
// vs 192MB L2), so the random gather + f32 atomic RMW resolves on-chip.
__global__ __launch_bounds__(256) void gcn_scatter(const int* __restrict__ src,
                                                   const int* __restrict__ dst,
                                                   const float* __restrict__ dinv,
                                                   const float* __restrict__ h,
                                                   float* __restrict__ out, int e) {
  int ed = blockIdx.x * 8 + (threadIdx.x >> 5);
  if (ed >= e) return;
  int lane = threadIdx.x & 31;
  int s = src[ed];
  int d = dst[ed];
  float c = dinv[s] * dinv[d];
  float4 hv = *(const float4*)&h[(size_t)s * D + lane * 4];
  float* op = &out[(size_t)d * D + lane * 4];
  unsafeAtomicAdd(op + 0, c * hv.x);  // -> global_atomic_add_f32
  unsafeAtomicAdd(op + 1, c * hv.y);
  unsafeAtomicAdd(op + 2, c * hv.z);
  unsafeAtomicAdd(op + 3, c * hv.w);
}

// ---------------- launch ----------------
extern "C" void kernel_launch(void* const* d_in, const int* in_sizes, int n_in,
                              void* d_out, int out_size, void* d_ws, size_t ws_size,
                              hipStream_t stream) {
  const float* x    = (const float*)d_in[0];
  const int*   edge = (const int*)d_in[1];  // [2,E]: row0=src, row1=dst (int32 per harness)
  const float* W    = (const float*)d_in[2];
  const float* b    = (const float*)d_in[3];
  float* out = (float*)d_out;

  const int n = in_sizes[0] / D;
  const int e = in_sizes[1] / 2;
  const int* src = edge;
  const int* dst = edge + e;

  // workspace layout: h [n*128 f32] | deg [n i32] | dinv [n f32]
  char* ws = (char*)d_ws;
  float* h   = (float*)ws;
  int*   deg = (int*)(ws + (size_t)n * D * sizeof(float));
  float* dinv = (float*)(ws + (size_t)n * D * sizeof(float) + (size_t)n * sizeof(int));

  gcn_deg_init <<<(n + 255) / 256, 256, 0, stream>>>(deg, n);
  gcn_deg_accum<<<(e + 255) / 256, 256, 0, stream>>>(dst, deg, e);
  gcn_dinv     <<<(n + 255) / 256, 256, 0, stream>>>(deg, dinv, n);
  gcn_gemm     <<<(n + 15) / 16, 256, 0, stream>>>(x, W, h, n);
  gcn_out_init <<<(int)(((size_t)n * 32 + 255) / 256), 256, 0, stream>>>(h, dinv, b, out, n);
  gcn_scatter  <<<(e + 7) / 8, 256, 0, stream>>>(src, dst, dinv, h, out, e);
}